// GCN_9371618640565
// MI455X (gfx1250) — compile-verified
//
#include <hip/hip_runtime.h>
#include <hip/hip_bf16.h>

// 3-layer GCN (PyG GCNConv semantics) + virtual-edge max agg + linear residual
// + BatchNorm/ReLU, for MI455X (gfx1250, wave32, WMMA).
//
// Dense GEMMs: V_WMMA_F32_16X16X4_F32, B matrices staged in LDS (320KB/WGP),
// A tile cached in registers -> inner loop is ds_load + dual WMMA only.
// Sparse aggregation: wave-per-edge global_atomic_add_f32 scatter (L2-resident).

#define NN 50000
#define FIN 128

typedef __attribute__((ext_vector_type(2))) float v2f;
typedef __attribute__((ext_vector_type(8))) float v8f;

// ---------------------------------------------------------------- utilities
__global__ void k_fill(float* __restrict__ p, float v, int n) {
  int i = blockIdx.x * blockDim.x + threadIdx.x;
  if (i < n) p[i] = v;
}

__global__ void k_count_deg(const int* __restrict__ dst, float* __restrict__ deg, int E) {
  int e = blockIdx.x * blockDim.x + threadIdx.x;
  if (e < E) atomicAdd(&deg[dst[e]], 1.0f);
}

__global__ void k_rsqrt_inplace(float* __restrict__ p, int n) {
  int i = blockIdx.x * blockDim.x + threadIdx.x;
  if (i < n) p[i] = rsqrtf(p[i]);  // deg >= 1 always (self-loop)
}

// ------------------------------------------------------- dual WMMA fp32 GEMM
// C1 = A @ B1, C2 = A @ B2.  A: M x 128 row-major. B: 128 x NC row-major.
// Block = 8 waves. B1,B2 staged in LDS once per block; each wave owns M-tiles
// (grid-strided), caches its A tile in 32 v2f registers, then sweeps all
// N-tiles: 32 K-steps x (4 ds_load_b32 + 2 v_wmma_f32_16x16x4_f32).
template <int NC>
__global__ void __launch_bounds__(256)
k_gemm_dual_wmma(const float* __restrict__ A,
                 const float* __restrict__ B1,
                 const float* __restrict__ B2,
                 float* __restrict__ C1,
                 float* __restrict__ C2,
                 int tilesM) {
  __shared__ float LB1[FIN * NC];
  __shared__ float LB2[FIN * NC];

  const int tid = threadIdx.x;
  // cooperative staging of both weight matrices (float4)
  for (int i = tid * 4; i < FIN * NC; i += 256 * 4) {
    *(float4*)&LB1[i] = *(const float4*)&B1[i];
    *(float4*)&LB2[i] = *(const float4*)&B2[i];
  }
  __syncthreads();

  const int wave = tid >> 5;
  const int lane = tid & 31;
  const int ln = lane & 15;          // A: row in tile | B/C: col in tile
  const int kb = (lane >> 4) << 1;   // K sub-offset: 0 (lanes 0-15) / 2 (lanes 16-31)
  const int gw = blockIdx.x * 8 + wave;
  const int strideW = gridDim.x * 8;

  for (int tm = gw; tm < tilesM; tm += strideW) {   // wave-uniform loop
    const int row0 = tm << 4;
    // cache A tile: lane needs row (row0+ln), K = 4s+kb, 4s+kb+1 for s=0..31
    const float* __restrict__ arow = A + (size_t)(row0 + ln) * FIN + kb;
    v2f av[32];
#pragma unroll
    for (int s = 0; s < 32; ++s) {
      av[s].x = arow[4 * s];
      av[s].y = arow[4 * s + 1];
    }

    for (int tn = 0; tn < NC / 16; ++tn) {
      const int col0 = tn << 4;
      const float* lb1 = LB1 + kb * NC + col0 + ln;  // +4*NC floats per K-step
      const float* lb2 = LB2 + kb * NC + col0 + ln;
      v8f acc1 = {};
      v8f acc2 = {};
#pragma unroll
      for (int s = 0; s < 32; ++s) {
        v2f b1, b2;
        b1.x = lb1[4 * s * NC];
        b1.y = lb1[4 * s * NC + NC];
        b2.x = lb2[4 * s * NC];
        b2.y = lb2[4 * s * NC + NC];
        acc1 = __builtin_amdgcn_wmma_f32_16x16x4_f32(false, av[s], false, b1,
                                                     (short)0, acc1, false, false);
        acc2 = __builtin_amdgcn_wmma_f32_16x16x4_f32(false, av[s], false, b2,
                                                     (short)0, acc2, false, false);
      }
      // C/D layout: VGPR i -> row (i + 8*(lane>=16)), col = lane&15
      const int rbase = row0 + ((lane >> 4) << 3);
#pragma unroll
      for (int i = 0; i < 8; ++i) {
        C1[(size_t)(rbase + i) * NC + col0 + ln] = acc1[i];
        C2[(size_t)(rbase + i) * NC + col0 + ln] = acc2[i];
      }
    }
  }
}

// ----------------------------------------------- self-loop init of agg buffers
// agg[v][f] = h[v][f] * dinv[v]^2   (self-loop term of symmetric normalization)
__global__ void k_init_selfloop(const float* __restrict__ h,
                                const float* __restrict__ dm,
                                const float* __restrict__ dv,
                                float* __restrict__ aggm,
                                float* __restrict__ aggv,
                                int n, int F) {
  int i = blockIdx.x * blockDim.x + threadIdx.x;
  if (i >= n * F) return;
  int v = i / F;
  float hv = h[i];
  float a = dm[v], b = dv[v];
  aggm[i] = hv * a * a;
  aggv[i] = hv * b * b;
}

// ------------------------------------------------------ edge scatter-aggregate
// One wave per edge: 32 lanes each own F/32 contiguous features (b128/b64 gather).
template <int F>
__global__ void __launch_bounds__(256)
k_edge_agg(const int* __restrict__ src, const int* __restrict__ dst,
           const float* __restrict__ h, const float* __restrict__ dinv,
           float* __restrict__ out, int E) {
  const int e = blockIdx.x * (blockDim.x >> 5) + (threadIdx.x >> 5);
  if (e >= E) return;
  const int lane = threadIdx.x & 31;
  const int s = src[e], d = dst[e];
  const float w = dinv[s] * dinv[d];
  constexpr int FPL = F / 32;  // 4 (F=128) or 2 (F=64)
  const float* __restrict__ hp = h + (size_t)s * F + lane * FPL;
  float* __restrict__ op = out + (size_t)d * F + lane * FPL;
  float hv[FPL];
#pragma unroll
  for (int j = 0; j < FPL; ++j) hv[j] = hp[j];   // vectorizes to b128/b64
#pragma unroll
  for (int j = 0; j < FPL; ++j) atomicAdd(&op[j], hv[j] * w);
}

// ------------------------------------------------- combine: max + bias + resid
__global__ void k_combine(const float* __restrict__ aggm,
                          const float* __restrict__ aggv,
                          const float* __restrict__ resid,
                          const float* __restrict__ bias,  // nullptr if none
                          float* __restrict__ out, int n, int F) {
  int i = blockIdx.x * blockDim.x + threadIdx.x;
  if (i >= n * F) return;
  float v = fmaxf(aggm[i], aggv[i]) + resid[i];
  if (bias) v += bias[i % F];
  out[i] = v;
}

// ---------------------------------------------------------------- batch norm
__global__ void k_bn_stats(const float* __restrict__ h, float* __restrict__ stats,
                           int n, int F) {
  int f = threadIdx.x;  // blockDim.x == F
  int rows_per_block = (n + gridDim.x - 1) / gridDim.x;
  int r0 = blockIdx.x * rows_per_block;
  int r1 = min(n, r0 + rows_per_block);
  float s = 0.f, s2 = 0.f;
  for (int r = r0; r < r1; ++r) {
    float v = h[(size_t)r * F + f];
    s += v;
    s2 += v * v;
  }
  atomicAdd(&stats[f], s);
  atomicAdd(&stats[F + f], s2);
}

__global__ void k_bn_relu(const float* __restrict__ h, const float* __restrict__ stats,
                          const float* __restrict__ gamma, const float* __restrict__ beta,
                          float* __restrict__ out, int n, int F) {
  int i = blockIdx.x * blockDim.x + threadIdx.x;
  if (i >= n * F) return;
  int f = i % F;
  float inv_n = 1.0f / (float)n;
  float mu = stats[f] * inv_n;
  float var = stats[F + f] * inv_n - mu * mu;  // biased, matches reference
  float v = (h[i] - mu) * rsqrtf(var + 1e-5f) * gamma[f] + beta[f];
  out[i] = fmaxf(v, 0.0f);
}

// ------------------------------------------------------------------- launcher
static inline dim3 g1(int total, int block = 256) { return dim3((total + block - 1) / block); }

extern "C" void kernel_launch(void* const* d_in, const int* in_sizes, int n_in,
                              void* d_out, int out_size, void* d_ws, size_t ws_size,
                              hipStream_t stream) {
  const float* x    = (const float*)d_in[0];
  const int*   ei   = (const int*)d_in[1];   // (2, Em) row-major
  const int*   vei  = (const int*)d_in[2];   // (2, Ev) row-major
  const float* Ws[3] = {(const float*)d_in[3], (const float*)d_in[4], (const float*)d_in[5]};
  const float* b2   = (const float*)d_in[6];
  const float* Ls[3] = {(const float*)d_in[7], (const float*)d_in[8], (const float*)d_in[9]};
  const float* gs[2]  = {(const float*)d_in[10], (const float*)d_in[12]};
  const float* bes[2] = {(const float*)d_in[11], (const float*)d_in[13]};

  const int Em = in_sizes[1] / 2;
  const int Ev = in_sizes[2] / 2;
  const int* src_m = ei;        const int* dst_m = ei + Em;
  const int* src_v = vei;       const int* dst_v = vei + Ev;

  // workspace layout (floats)
  float* ws = (float*)d_ws;
  const size_t NF = (size_t)NN * FIN;
  float* xcur  = ws;            // layer output (next layer input)
  float* hbuf  = ws + 1 * NF;   // x@W, then combined pre-BN result
  float* resid = ws + 2 * NF;   // x@L
  float* aggm  = ws + 3 * NF;
  float* aggv  = ws + 4 * NF;
  float* dinvm = ws + 5 * NF;            // NN floats (deg then dinv, in place)
  float* dinvv = dinvm + NN;             // NN floats
  float* stats = dinvv + NN;             // 2*FIN floats

  // ---- degrees / dinv (shared by all layers) ----
  k_fill<<<g1(NN), 256, 0, stream>>>(dinvm, 1.0f, NN);  // self-loop
  k_fill<<<g1(NN), 256, 0, stream>>>(dinvv, 1.0f, NN);
  k_count_deg<<<g1(Em), 256, 0, stream>>>(dst_m, dinvm, Em);
  k_count_deg<<<g1(Ev), 256, 0, stream>>>(dst_v, dinvv, Ev);
  k_rsqrt_inplace<<<g1(NN), 256, 0, stream>>>(dinvm, NN);
  k_rsqrt_inplace<<<g1(NN), 256, 0, stream>>>(dinvv, NN);

  const int tilesM = NN / 16;  // 3125
  const float* xin = x;
  for (int layer = 0; layer < 3; ++layer) {
    const int Fout = (layer == 2) ? 64 : 128;
    const int elems = NN * Fout;

    // h = xin @ W, resid = xin @ L  (fused dual WMMA GEMM, B in LDS)
    if (Fout == 128) {
      k_gemm_dual_wmma<128><<<dim3(256), 256, 0, stream>>>(
          xin, Ws[layer], Ls[layer], hbuf, resid, tilesM);
    } else {
      k_gemm_dual_wmma<64><<<dim3(256), 256, 0, stream>>>(
          xin, Ws[layer], Ls[layer], hbuf, resid, tilesM);
    }

    // self-loop terms, then edge scatter (main + virtual graphs)
    k_init_selfloop<<<g1(elems), 256, 0, stream>>>(hbuf, dinvm, dinvv, aggm, aggv, NN, Fout);
    if (Fout == 128) {
      k_edge_agg<128><<<dim3((Em + 7) / 8), 256, 0, stream>>>(src_m, dst_m, hbuf, dinvm, aggm, Em);
      k_edge_agg<128><<<dim3((Ev + 7) / 8), 256, 0, stream>>>(src_v, dst_v, hbuf, dinvv, aggv, Ev);
    } else {
      k_edge_agg<64><<<dim3((Em + 7) / 8), 256, 0, stream>>>(src_m, dst_m, hbuf, dinvm, aggm, Em);
      k_edge_agg<64><<<dim3((Ev + 7) / 8), 256, 0, stream>>>(src_v, dst_v, hbuf, dinvv, aggv, Ev);
    }

    // h = max(agg_main, agg_virt) (+bias on last layer) + residual
    float* comb_out = (layer == 2) ? (float*)d_out : hbuf;
    const float* bias = (layer == 2) ? b2 : nullptr;
    k_combine<<<g1(elems), 256, 0, stream>>>(aggm, aggv, resid, bias, comb_out, NN, Fout);

    if (layer < 2) {
      k_fill<<<1, 256, 0, stream>>>(stats, 0.0f, 2 * FIN);
      k_bn_stats<<<dim3(256), FIN, 0, stream>>>(hbuf, stats, NN, FIN);
      k_bn_relu<<<g1(elems), 256, 0, stream>>>(hbuf, stats, gs[layer], bes[layer], xcur, NN, FIN);
      xin = xcur;
    }
  }
}